// DGMANMSCenterExtractor_54606214201836
// MI455X (gfx1250) — compile-verified
//
#include <hip/hip_runtime.h>
#include <hip/hip_bf16.h>
#include <float.h>
#include <stdint.h>

// ---------------------------------------------------------------------------
// NMS center extractor: 3x3 local-max + per-image top-5 over (64,512,512) fp32.
// Memory-bound (64 MiB in, ~2.9 us floor @ 23.3 TB/s). Data movement uses the
// CDNA5 Tensor Data Mover (tensor_load_to_lds + s_wait_tensorcnt); compute
// reads the staged tile with vectorized ds_load_b128 row segments.
// ---------------------------------------------------------------------------

#define H 512
#define W 512
#define NIMG 64
#define BPI 16          // blocks per image
#define TROWS (H / BPI) // compute rows per block = 32 (single LDS tile)
#define NTHREADS 256
#define KTOP 5
#define SEGS (TROWS * W / 8)  // 8-pixel segments per tile = 2048

typedef unsigned int v4u __attribute__((ext_vector_type(4)));
typedef int          v4i __attribute__((ext_vector_type(4)));
typedef int          v8i __attribute__((ext_vector_type(8)));

// Issue a 2D TDM tile load: `rows` x `width_elems` fp32 tile (row stride
// `stride_elems`) from global `gaddr` into LDS byte offset `lds_off`.
// D# layout per CDNA5 ISA (groups 0/1; groups 2/3 NULL
// for a 2D tensor). Must be executed by a single wave.
__device__ __forceinline__ void tdm_load_2d_f32(uint32_t lds_off, const void* gaddr,
                                                uint32_t width_elems, uint32_t rows,
                                                uint32_t stride_elems) {
    uint64_t ga = (uint64_t)(uintptr_t)gaddr;
    v4u g0;
    g0.x = 1u;                                      // count=1 (valid), user mode, no gather
    g0.y = lds_off;                                 // lds_addr
    g0.z = (uint32_t)ga;                            // global_addr[31:0]
    g0.w = (uint32_t)((ga >> 32) & 0x1FFFFFFu)      // global_addr[56:32]
         | (2u << 30);                              // type = 2 ("image")
    v8i g1;
    g1[0] = (int)(2u << 16);                        // wg_mask=0, data_size=2 (4B), no flags
    g1[1] = (int)((width_elems & 0xFFFFu) << 16);   // abar_addr=0 | tensor_dim0[15:0]
    g1[2] = (int)(((width_elems >> 16) & 0xFFFFu)   // tensor_dim0[31:16]
         | ((rows & 0xFFFFu) << 16));               // tensor_dim1[15:0]
    g1[3] = (int)(((rows >> 16) & 0xFFFFu)          // tensor_dim1[31:16]
         | ((width_elems & 0xFFFFu) << 16));        // tile_dim0 = width
    g1[4] = (int)(rows & 0xFFFFu);                  // tile_dim1 = rows, tile_dim2 = 0
    g1[5] = (int)stride_elems;                      // tensor_dim0_stride[31:0]
    g1[6] = 0;                                      // stride0[47:32], stride1[15:0]
    g1[7] = 0;                                      // stride1[47:16]
    v4i z4 = {0, 0, 0, 0};
#if __clang_major__ >= 23
    v8i z8 = {0, 0, 0, 0, 0, 0, 0, 0};
    __builtin_amdgcn_tensor_load_to_lds(g0, g1, z4, z4, z8, 0);
#else
    __builtin_amdgcn_tensor_load_to_lds(g0, g1, z4, z4, 0);
#endif
}

// Monotonic composite key: heatmap values are uniform [0,1) >= 0, so the raw
// fp32 bit pattern preserves order; ~index makes equal values prefer the
// smaller index (jax.lax.top_k tie-break).
__device__ __forceinline__ unsigned long long make_key(float v, unsigned idx) {
    return ((unsigned long long)__float_as_uint(v) << 32) | (unsigned)(~idx);
}

// Load 10 contiguous row values [x0-1, x0+8] from an LDS row into arr[0..9],
// using two ds_load_b128 for the aligned core and -FLT_MAX at image borders.
__device__ __forceinline__ void load_row10(const float* __restrict__ rowp, int x0,
                                           float* __restrict__ arr) {
    const float4 a = *(const float4*)(rowp + x0);       // x0 is a multiple of 8
    const float4 b = *(const float4*)(rowp + x0 + 4);
    arr[0] = (x0 > 0) ? rowp[x0 - 1] : -FLT_MAX;
    arr[1] = a.x; arr[2] = a.y; arr[3] = a.z; arr[4] = a.w;
    arr[5] = b.x; arr[6] = b.y; arr[7] = b.z; arr[8] = b.w;
    arr[9] = (x0 + 8 < W) ? rowp[x0 + 8] : -FLT_MAX;
}

__device__ __forceinline__ void fill_neg10(float* __restrict__ arr) {
    #pragma unroll
    for (int j = 0; j < 10; ++j) arr[j] = -FLT_MAX;
}

__global__ void __launch_bounds__(NTHREADS)
nms_topk_stage1(const float* __restrict__ hm, unsigned long long* __restrict__ blockKeys) {
    __shared__ float tile[(TROWS + 2) * W];                 // 34*512*4 = 69632 B
    __shared__ unsigned long long red[NTHREADS];            // 2 KiB

    const int img  = blockIdx.x / BPI;
    const int sub  = blockIdx.x % BPI;
    const int row0 = sub * TROWS;
    const float* base = hm + (size_t)img * H * W;
    const int tid = threadIdx.x;

    // One TDM tile per block: rows [row0-1, row0+TROWS+1) clamped to image.
    const int g0r   = (row0 - 1 < 0) ? 0 : row0 - 1;
    const int gend  = (row0 + TROWS + 1 > H) ? H : row0 + TROWS + 1;
    const int nrows = gend - g0r;                           // <= TROWS+2
    if (tid < 32) {                                         // single wave drives the DMA
        tdm_load_2d_f32((uint32_t)(size_t)tile, base + (size_t)g0r * W,
                        W, (uint32_t)nrows, W);
        __builtin_amdgcn_s_wait_tensorcnt(0);
    }
    __syncthreads();                                        // LDS tile visible to all waves

    unsigned long long top[KTOP] = {0ull, 0ull, 0ull, 0ull, 0ull};

    // Each thread processes 8 segments of 8 contiguous pixels: 3 rows of
    // 10 values each (2x ds_load_b128 + 2 edge scalars per row), all 3x3
    // neighbor maxes computed on registers.
    for (int s = tid; s < SEGS; s += NTHREADS) {
        const int ry = s >> 6;                              // row within tile, 0..31
        const int x0 = (s & 63) << 3;                       // segment start column
        const int y  = row0 + ry;
        const int ly = y - g0r;

        float r0[10], r1[10], r2[10];
        if (y > 0)     load_row10(&tile[(ly - 1) * W], x0, r0); else fill_neg10(r0);
                       load_row10(&tile[ ly      * W], x0, r1);
        if (y < H - 1) load_row10(&tile[(ly + 1) * W], x0, r2); else fill_neg10(r2);

        #pragma unroll
        for (int j = 0; j < 8; ++j) {
            const float v = r1[j + 1];
            const float m = fmaxf(fmaxf(fmaxf(r0[j], r0[j + 1]), fmaxf(r0[j + 2], r1[j])),
                                  fmaxf(fmaxf(r1[j + 2], r2[j]), fmaxf(r2[j + 1], r2[j + 2])));
            if (v >= m) {                                   // pooled == h  (ties survive)
                const unsigned long long key = make_key(v, (unsigned)(y * W + x0 + j));
                if (key > top[KTOP - 1]) {                  // sorted-desc insert
                    top[KTOP - 1] = key;
                    #pragma unroll
                    for (int k = KTOP - 1; k > 0; --k) {
                        if (top[k] > top[k - 1]) {
                            unsigned long long tmp = top[k];
                            top[k] = top[k - 1];
                            top[k - 1] = tmp;
                        }
                    }
                }
            }
        }
    }
    __syncthreads();

    // Block-wide top-5: 5 rounds of u64 max-reduce; winner's owner pops its slot.
    // Keys are globally unique (each embeds a distinct pixel index), so at most
    // one thread matches the winner, and it is always that thread's top[0].
    for (int r = 0; r < KTOP; ++r) {
        red[tid] = top[0];
        __syncthreads();
        for (int s = NTHREADS / 2; s > 0; s >>= 1) {
            if (tid < s) {
                if (red[tid + s] > red[tid]) red[tid] = red[tid + s];
            }
            __syncthreads();
        }
        const unsigned long long win = red[0];
        __syncthreads();
        if (win != 0ull && top[0] == win) {
            #pragma unroll
            for (int k = 0; k < KTOP - 1; ++k) top[k] = top[k + 1];
            top[KTOP - 1] = 0ull;
        }
        if (tid == 0) blockKeys[(size_t)blockIdx.x * KTOP + r] = win;
    }
}

__global__ void __launch_bounds__(128)
nms_topk_stage2(const unsigned long long* __restrict__ blockKeys, float* __restrict__ out) {
    __shared__ unsigned long long keys[BPI * KTOP];         // 80 keys per image
    const int img = blockIdx.x;
    const int tid = threadIdx.x;
    if (tid < BPI * KTOP) keys[tid] = blockKeys[(size_t)img * BPI * KTOP + tid];
    __syncthreads();
    if (tid == 0) {
        for (int r = 0; r < KTOP; ++r) {
            int best = 0;
            for (int j = 1; j < BPI * KTOP; ++j)
                if (keys[j] > keys[best]) best = j;
            const unsigned long long k = keys[best];
            keys[best] = 0ull;
            const float v = __uint_as_float((unsigned)(k >> 32));
            const unsigned idx = ~((unsigned)k);            // k==0 -> invalid, masked below
            const float valid = (v >= 0.3f) ? 1.0f : 0.0f;
            const float rowf = (float)(idx / W);
            const float colf = (float)(idx % W);
            const float nx = (2.0f * colf / (float)(W - 1) - 1.0f) * valid;
            const float ny = (2.0f * rowf / (float)(H - 1) - 1.0f) * valid;
            // return-order concat: centers (64,5,2) | valid (64,5) | top_vals (64,5)
            out[(size_t)img * (KTOP * 2) + r * 2 + 0] = nx;
            out[(size_t)img * (KTOP * 2) + r * 2 + 1] = ny;
            out[NIMG * KTOP * 2 + (size_t)img * KTOP + r] = valid;
            out[NIMG * KTOP * 3 + (size_t)img * KTOP + r] = v;
        }
    }
}

extern "C" void kernel_launch(void* const* d_in, const int* in_sizes, int n_in,
                              void* d_out, int out_size, void* d_ws, size_t ws_size,
                              hipStream_t stream) {
    (void)in_sizes; (void)n_in; (void)out_size; (void)ws_size;
    const float* hm = (const float*)d_in[0];
    unsigned long long* blockKeys = (unsigned long long*)d_ws;  // 1024*5*8 = 40 KiB
    float* out = (float*)d_out;

    nms_topk_stage1<<<NIMG * BPI, NTHREADS, 0, stream>>>(hm, blockKeys);
    nms_topk_stage2<<<NIMG, 128, 0, stream>>>(blockKeys, out);
}